// CRF_16149077033429
// MI455X (gfx1250) — compile-verified
//
#include <hip/hip_runtime.h>

// ---------------- problem constants ----------------
#define T50    50     // tag count
#define TP     64     // padded tag count (WMMA tiles)
#define STARTT 48
#define STOPT  49
#define NB     64     // batch
#define SQ     512    // sequence length
#define ROWS   16     // batch rows per workgroup (WMMA M)
#define PSTR   68     // padded LDS row stride in floats -> conflict-free frag loads

typedef __attribute__((ext_vector_type(2))) float v2f;
typedef __attribute__((ext_vector_type(4))) float v4f;
typedef __attribute__((ext_vector_type(8))) float v8f;

// ---------------- LDS layout (float offsets) ----------------
// Epack : exp(transitions) pre-swizzled into WMMA B-fragment order
//         64 fragments (k-chunk*4 + n-tile) * 32 lanes * 2 floats
#define EP_OFF 0
#define P_OFF  (EP_OFF + 64*32*2)     // partition  16 x PSTR
#define F_OFF  (P_OFF  + ROWS*PSTR)   // feats tile 16 x PSTR
#define MX_OFF (F_OFF  + ROWS*PSTR)   // per-row max, 16
#define MK_OFF (MX_OFF + 16)          // per-row mask, 16
#define LDS_FLOATS (MK_OFF + 16)      // ~25 KB of 320 KB/WGP

#define NEG_INF (-__builtin_inff())

// =====================================================================
// Forward algorithm: one wave32 handles 16 batch rows; per time step one
// 16x64x64 f32 GEMM in exp-space via V_WMMA_F32_16X16X4_F32.
// k-outer / n-inner ordering keeps FOUR independent accumulator chains
// live, so the 16-deep WMMA RAW chains of the 4 N-tiles overlap --
// per-step latency is what bounds this 512-step serial scan.
// =====================================================================
__global__ __launch_bounds__(32)
void crf_forward(const float* __restrict__ feats, const int* __restrict__ mask,
                 const float* __restrict__ trans, float* __restrict__ ws_fwd) {
  __shared__ float lds[LDS_FLOATS];
  const int lane = threadIdx.x;          // 0..31 (wave32)
  const int b0   = blockIdx.x * ROWS;
  const int hi   = (lane >= 16) ? 1 : 0; // which lane half
  const int l15  = lane & 15;

  // ---- build Epack = exp(transitions), already in B-fragment layout ----
  // B 4x16 fragment (k,n): VGPR0 = E[4k + 2*hi][16n + l15], VGPR1 = next K row.
  for (int f = 0; f < 64; ++f) {
    const int k = f >> 2, n = f & 3;
    const int ki = 4*k + 2*hi;
    const int j  = 16*n + l15;
    float e0 = 0.f, e1 = 0.f;
    if (j < T50) {
      if (ki     < T50) e0 = __expf(trans[ki*T50 + j]);
      if (ki + 1 < T50) e1 = __expf(trans[(ki+1)*T50 + j]);
    }
    lds[EP_OFF + (f*32 + lane)*2 + 0] = e0;
    lds[EP_OFF + (f*32 + lane)*2 + 1] = e1;
  }

  // ---- t = 0: partition[b,j] = feats[b,0,j] + trans[START,j]; pad = -inf ----
  for (int idx = lane; idx < ROWS*TP; idx += 32) {
    const int r = idx >> 6, j = idx & 63;
    float v = NEG_INF;
    if (j < T50) v = feats[((size_t)(b0 + r)*SQ + 0)*T50 + j] + trans[STARTT*T50 + j];
    lds[P_OFF + r*PSTR + j] = v;
  }
  __syncthreads();

  // feats prefetch: idx = lane + 32*i covers a 16x64 tile; i even -> j=lane,
  // i odd -> j=lane+32 (valid only for lane<18 since T=50).
  float freg[32];
  auto prefetch = [&](int t) {
    #pragma unroll
    for (int i = 0; i < 32; ++i) {
      const int r = i >> 1;
      const int j = (i & 1) ? (32 + lane) : lane;
      freg[i] = (j < T50) ? feats[((size_t)(b0 + r)*SQ + t)*T50 + j] : 0.f;
    }
  };
  prefetch(1);

  for (int t = 1; t < SQ; ++t) {
    // stage this step's feats tile from registers into LDS
    #pragma unroll
    for (int i = 0; i < 32; ++i) {
      const int r = i >> 1;
      const int j = (i & 1) ? (32 + lane) : lane;
      lds[F_OFF + r*PSTR + j] = freg[i];
    }
    if (lane < 16) lds[MK_OFF + lane] = (float)mask[(b0 + lane)*SQ + t];
    // issue next step's global loads now; they complete under the compute below
    if (t + 1 < SQ) prefetch(t + 1);

    // ---- per-row max of partition (lane halves split the 64 cols) ----
    float h = NEG_INF;
    {
      const int cbase = hi * 32;
      #pragma unroll
      for (int c = 0; c < 8; ++c) {
        v4f p = *(const v4f*)&lds[P_OFF + l15*PSTR + cbase + c*4];
        h = fmaxf(h, fmaxf(fmaxf(p.x, p.y), fmaxf(p.z, p.w)));
      }
    }
    const float mx = fmaxf(h, __shfl_xor(h, 16, 32));
    if (lane < 16) lds[MX_OFF + lane] = mx;

    // ---- A fragments: exp(partition - rowmax), ISA 16x4 f32 A layout ----
    v2f a[16];
    #pragma unroll
    for (int k = 0; k < 16; ++k) {
      v2f p = *(const v2f*)&lds[P_OFF + l15*PSTR + 4*k + 2*hi];
      a[k].x = __expf(p.x - mx);
      a[k].y = __expf(p.y - mx);
    }
    __syncthreads();

    // ---- S = A(16x64) * E(64x64): 4 interleaved accumulator chains ----
    v8f acc[4];
    #pragma unroll
    for (int n = 0; n < 4; ++n)
      acc[n] = (v8f){0.f, 0.f, 0.f, 0.f, 0.f, 0.f, 0.f, 0.f};
    #pragma unroll
    for (int k = 0; k < 16; ++k) {
      #pragma unroll
      for (int n = 0; n < 4; ++n) {
        v2f b = *(const v2f*)&lds[EP_OFF + ((k*4 + n)*32 + lane)*2];
        acc[n] = __builtin_amdgcn_wmma_f32_16x16x4_f32(
                     false, a[k], false, b, (short)0, acc[n], false, false);
      }
    }

    // ---- partition update: P' = mx + log(S) + feats, masked select ----
    // D layout: VGPR r -> row (r + 8*hi), col 16n + l15
    #pragma unroll
    for (int n = 0; n < 4; ++n) {
      const int j = 16*n + l15;
      if (j < T50) {
        #pragma unroll
        for (int r = 0; r < 8; ++r) {
          const int m      = r + 8*hi;
          const float cur  = lds[MX_OFF + m] + __logf(acc[n][r]) + lds[F_OFF + m*PSTR + j];
          const float oldv = lds[P_OFF + m*PSTR + j];
          const float mk   = lds[MK_OFF + m];
          lds[P_OFF + m*PSTR + j] = (mk != 0.f) ? cur : oldv;
        }
      }
    }
    __syncthreads();
  }

  // ---- terminal: logsumexp_i(partition[b,i] + trans[i,STOP]) per row ----
  float term = 0.f;
  if (lane < 16) {
    const int m = lane;
    float mxv = NEG_INF;
    for (int i = 0; i < T50; ++i)
      mxv = fmaxf(mxv, lds[P_OFF + m*PSTR + i] + trans[i*T50 + STOPT]);
    float s = 0.f;
    for (int i = 0; i < T50; ++i)
      s += __expf(lds[P_OFF + m*PSTR + i] + trans[i*T50 + STOPT] - mxv);
    term = mxv + __logf(s);
  }
  #pragma unroll
  for (int off = 16; off > 0; off >>= 1) term += __shfl_xor(term, off, 32);
  if (lane == 0) ws_fwd[blockIdx.x] = term;
}

// =====================================================================
// Gold score: one block per batch element, 64 threads stride the sequence.
// =====================================================================
__global__ __launch_bounds__(64)
void crf_gold(const float* __restrict__ feats, const int* __restrict__ mask,
              const int* __restrict__ tags, const float* __restrict__ trans,
              float* __restrict__ ws_gold) {
  __shared__ float redf[64];
  __shared__ int   redi[64];
  const int b = blockIdx.x, tid = threadIdx.x;
  float acc = 0.f; int lenacc = 0;
  for (int s = tid; s < SQ; s += 64) {
    const int tg   = tags[b*SQ + s];
    const int mk   = mask[b*SQ + s];
    const int prev = (s == 0) ? STARTT : tags[b*SQ + s - 1];
    const float e  = feats[((size_t)b*SQ + s)*T50 + tg];
    const float tr = trans[prev*T50 + tg];
    acc    += (e + tr) * (float)mk;
    lenacc += mk;
  }
  redf[tid] = acc; redi[tid] = lenacc;
  __syncthreads();
  for (int off = 32; off > 0; off >>= 1) {
    if (tid < off) { redf[tid] += redf[tid + off]; redi[tid] += redi[tid + off]; }
    __syncthreads();
  }
  if (tid == 0) {
    const int len   = redi[0];
    const int endid = tags[b*SQ + (len > 0 ? len - 1 : 0)];
    ws_gold[b] = redf[0] + trans[endid*T50 + STOPT];
  }
}

// =====================================================================
// Combine: out = sum(forward partials) - sum(gold partials)
// =====================================================================
__global__ void crf_combine(const float* __restrict__ ws, float* __restrict__ out) {
  if (threadIdx.x == 0 && blockIdx.x == 0) {
    float fwd = 0.f, gold = 0.f;
    for (int i = 0; i < 4;  ++i) fwd  += ws[i];
    for (int i = 0; i < NB; ++i) gold += ws[4 + i];
    out[0] = fwd - gold;
  }
}

extern "C" void kernel_launch(void* const* d_in, const int* in_sizes, int n_in,
                              void* d_out, int out_size, void* d_ws, size_t ws_size,
                              hipStream_t stream) {
  const float* feats = (const float*)d_in[0];   // (64,512,50) f32
  const int*   mask  = (const int*)  d_in[1];   // (64,512)    i32
  const int*   tags  = (const int*)  d_in[2];   // (64,512)    i32
  const float* trans = (const float*)d_in[3];   // (50,50)     f32
  float* ws = (float*)d_ws;                     // [0:4) fwd partials, [4:68) gold

  crf_forward<<<4, 32, 0, stream>>>(feats, mask, trans, ws);
  crf_gold   <<<NB, 64, 0, stream>>>(feats, mask, tags, trans, ws + 4);
  crf_combine<<<1, 32, 0, stream>>>(ws, (float*)d_out);
}